// SignedDistanceFieldModule_22265110462642
// MI455X (gfx1250) — compile-verified
//
#include <hip/hip_runtime.h>
#include <hip/hip_bf16.h>

typedef __attribute__((ext_vector_type(2))) float v2f;
typedef __attribute__((ext_vector_type(8))) float v8f;

#define BLOCK 256
#define CHUNK 1024   // triangles staged in LDS at a time (1024*9*4 = 36 KB)

__global__ __launch_bounds__(BLOCK)
void sdf_wmma_kernel(const float* __restrict__ points,
                     const float* __restrict__ verts,
                     const int*   __restrict__ tidx,
                     float*       __restrict__ out,
                     int nPts, int nTri)
{
    __shared__ float tsm[CHUNK * 9];

    const int tid  = threadIdx.x;
    const int lane = tid & 31;
    const int wave = tid >> 5;
    const int h    = lane >> 4;     // half-wave
    const int n16  = lane & 15;     // triangle column in tile / point row for A operand
    const int ptBase = ((int)blockIdx.x * (BLOCK / 32) + wave) * 16;

    // ---- WMMA A operand: 16x4 f32 points matrix (K=3 used, pad 0) ----
    // lanes 0-15 hold (K0,K1)=(x,y); lanes 16-31 hold (K2,K3)=(z,0)
    v2f amat;
    {
        int pidx = ptBase + n16; if (pidx >= nPts) pidx = nPts - 1;
        const float* P = points + 3 * pidx;
        amat.x = h ? P[2] : P[0];
        amat.y = h ? 0.0f : P[1];
    }

    // ---- per-slot point data: slot s of the 16x16 C tile maps to M = s + 8*h ----
    float px[8], py[8], pz[8], pp[8];
#pragma unroll
    for (int s = 0; s < 8; ++s) {
        int pidx = ptBase + s + 8 * h; if (pidx >= nPts) pidx = nPts - 1;
        const float* P = points + 3 * pidx;
        px[s] = P[0]; py[s] = P[1]; pz[s] = P[2];
        pp[s] = px[s]*px[s] + py[s]*py[s] + pz[s]*pz[s];
    }

    float bestD2[8], bu[8], bv[8], bw[8], osum[8];
    int   bface[8];
#pragma unroll
    for (int s = 0; s < 8; ++s) {
        bestD2[s] = 3.402823466e38f;
        bu[s] = 1.f; bv[s] = 0.f; bw[s] = 0.f;
        bface[s] = 0; osum[s] = 0.f;
    }

    const float eps = 1e-12f;

    for (int cs = 0; cs < nTri; cs += CHUNK) {
        int lim = nTri - cs; if (lim > CHUNK) lim = CHUNK;
        __syncthreads();
        for (int t = tid; t < lim; t += BLOCK) {
            int g  = cs + t;
            int i0 = tidx[3*g], i1 = tidx[3*g+1], i2 = tidx[3*g+2];
            float* s = tsm + t * 9;
            s[0]=verts[3*i0]; s[1]=verts[3*i0+1]; s[2]=verts[3*i0+2];
            s[3]=verts[3*i1]; s[4]=verts[3*i1+1]; s[5]=verts[3*i1+2];
            s[6]=verts[3*i2]; s[7]=verts[3*i2+1]; s[8]=verts[3*i2+2];
        }
        __syncthreads();

        int nTiles = (lim + 15) >> 4;
        for (int j = 0; j < nTiles; ++j) {
            const int  triIdx = cs + j * 16 + n16;
            const bool tvalid = triIdx < nTri;
            const float* T = tsm + (j * 16 + n16) * 9;
            float Ax=T[0],Ay=T[1],Az=T[2];
            float Bx=T[3],By=T[4],Bz=T[5];
            float Cx=T[6],Cy=T[7],Cz=T[8];
            float abx=Bx-Ax, aby=By-Ay, abz=Bz-Az;
            float acx=Cx-Ax, acy=Cy-Ay, acz=Cz-Az;

            // per-triangle scalars (same for all 8 slots of this lane)
            float abA=abx*Ax+aby*Ay+abz*Az;
            float acA=acx*Ax+acy*Ay+acz*Az;
            float abB=abx*Bx+aby*By+abz*Bz;
            float acB=acx*Bx+acy*By+acz*Bz;
            float abC=abx*Cx+aby*Cy+abz*Cz;
            float acC=acx*Cx+acy*Cy+acz*Cz;
            float aa =Ax*Ax+Ay*Ay+Az*Az;
            float bb =Bx*Bx+By*By+Bz*Bz;
            float cc =Cx*Cx+Cy*Cy+Cz*Cz;
            float abd=Ax*Bx+Ay*By+Az*Bz;
            float acd=Ax*Cx+Ay*Cy+Az*Cz;
            float bcd=Bx*Cx+By*Cy+Bz*Cz;

            // ---- B operands: 4x16 f32. V0 carries K0(low half)/K2(high half),
            //      V1 carries K1(low)/K3=0(high). ----
            v2f bAB; bAB.x = h ? abz : abx; bAB.y = h ? 0.f : aby;
            v2f bAC; bAC.x = h ? acz : acx; bAC.y = h ? 0.f : acy;
            v2f bA;  bA.x  = h ? Az  : Ax;  bA.y  = h ? 0.f : Ay;
            v2f bB;  bB.x  = h ? Bz  : Bx;  bB.y  = h ? 0.f : By;
            v2f bC;  bC.x  = h ? Cz  : Cx;  bC.y  = h ? 0.f : Cy;

            v8f zc = {0.f,0.f,0.f,0.f,0.f,0.f,0.f,0.f};
            // D[M][N] = dot(point_M, vec_N): 5 planes of the 16x16 tile
            v8f Dab = __builtin_amdgcn_wmma_f32_16x16x4_f32(false, amat, false, bAB, (short)0, zc, false, false);
            v8f Dac = __builtin_amdgcn_wmma_f32_16x16x4_f32(false, amat, false, bAC, (short)0, zc, false, false);
            v8f DA  = __builtin_amdgcn_wmma_f32_16x16x4_f32(false, amat, false, bA,  (short)0, zc, false, false);
            v8f DB  = __builtin_amdgcn_wmma_f32_16x16x4_f32(false, amat, false, bB,  (short)0, zc, false, false);
            v8f DC  = __builtin_amdgcn_wmma_f32_16x16x4_f32(false, amat, false, bC,  (short)0, zc, false, false);

#pragma unroll
            for (int s = 0; s < 8; ++s) {
                float pab = Dab[s], pac = Dac[s];
                float pa  = DA[s],  pb  = DB[s], pc = DC[s];
                float d1 = pab - abA, d2 = pac - acA;
                float d3 = pab - abB, d4 = pac - acB;
                float d5 = pab - abC, d6 = pac - acC;
                float vc_ = d1*d4 - d3*d2;
                float vb_ = d5*d2 - d1*d6;
                float va_ = d3*d6 - d5*d4;

                float den = fmaxf(va_ + vb_ + vc_, eps);
                float v = vb_ / den, w = vc_ / den, u = 1.f - v - w;

                float e43 = d4 - d3, e56 = d5 - d6;
                float tbc = e43 / fmaxf(e43 + e56, eps);
                bool  c = (va_ <= 0.f) && (e43 >= 0.f) && (e56 >= 0.f);
                u = c ? 0.f        : u;  v = c ? (1.f - tbc) : v;  w = c ? tbc : w;

                float tac = d2 / fmaxf(d2 - d6, eps);
                c = (vb_ <= 0.f) && (d2 >= 0.f) && (d6 <= 0.f);
                u = c ? (1.f - tac) : u;  v = c ? 0.f : v;  w = c ? tac : w;

                float tab = d1 / fmaxf(d1 - d3, eps);
                c = (vc_ <= 0.f) && (d1 >= 0.f) && (d3 <= 0.f);
                u = c ? (1.f - tab) : u;  v = c ? tab : v;  w = c ? 0.f : w;

                c = (d6 >= 0.f) && (d5 <= d6);
                u = c ? 0.f : u;  v = c ? 0.f : v;  w = c ? 1.f : w;
                c = (d3 >= 0.f) && (d4 <= d3);
                u = c ? 0.f : u;  v = c ? 1.f : v;  w = c ? 0.f : w;
                c = (d1 <= 0.f) && (d2 <= 0.f);
                u = c ? 1.f : u;  v = c ? 0.f : v;  w = c ? 0.f : w;

                float qp = u*pa + v*pb + w*pc;
                float qq = u*u*aa + v*v*bb + w*w*cc
                         + 2.f*(u*v*abd + u*w*acd + v*w*bcd);
                float dist2 = fmaxf(pp[s] - 2.f*qp + qq, 0.f);

                bool upd = tvalid && (dist2 < bestD2[s]);
                bestD2[s] = upd ? dist2  : bestD2[s];
                bu[s]     = upd ? u      : bu[s];
                bv[s]     = upd ? v      : bv[s];
                bw[s]     = upd ? w      : bw[s];
                bface[s]  = upd ? triIdx : bface[s];

                // winding-number solid angle for this (point, triangle) pair
                float rax = Ax - px[s], ray = Ay - py[s], raz = Az - pz[s];
                float rbx = Bx - px[s], rby = By - py[s], rbz = Bz - pz[s];
                float rcx = Cx - px[s], rcy = Cy - py[s], rcz = Cz - pz[s];
                float la = sqrtf(rax*rax + ray*ray + raz*raz);
                float lb = sqrtf(rbx*rbx + rby*rby + rbz*rbz);
                float lc = sqrtf(rcx*rcx + rcy*rcy + rcz*rcz);
                float cxx = rby*rcz - rbz*rcy;
                float cxy = rbz*rcx - rbx*rcz;
                float cxz = rbx*rcy - rby*rcx;
                float det  = rax*cxx + ray*cxy + raz*cxz;
                float dab_ = rax*rbx + ray*rby + raz*rbz;
                float dbc_ = rbx*rcx + rby*rcy + rbz*rcz;
                float dca_ = rcx*rax + rcy*ray + rcz*raz;
                float denw = la*lb*lc + dab_*lc + dbc_*la + dca_*lb;
                float om   = 2.f * atan2f(det, denw);
                osum[s] += tvalid ? om : 0.f;
            }
        }
    }

    // ---- reduce over the 16 triangle columns (lanes within each half-wave) ----
#pragma unroll
    for (int s = 0; s < 8; ++s) {
        float d2v = bestD2[s], u = bu[s], v = bv[s], w = bw[s], om = osum[s];
        int   f   = bface[s];
#pragma unroll
        for (int m = 1; m < 16; m <<= 1) {
            float od2 = __shfl_xor(d2v, m);
            float ou  = __shfl_xor(u,   m);
            float ov  = __shfl_xor(v,   m);
            float ow  = __shfl_xor(w,   m);
            int   of  = __shfl_xor(f,   m);
            om       += __shfl_xor(om,  m);
            bool take = (od2 < d2v) || ((od2 == d2v) && (of < f));
            d2v = take ? od2 : d2v;
            u   = take ? ou  : u;
            v   = take ? ov  : v;
            w   = take ? ow  : w;
            f   = take ? of  : f;
        }
        if (n16 == 0) {
            int gp = ptBase + s + 8 * h;
            if (gp < nPts) {
                int i0 = tidx[3*f], i1 = tidx[3*f+1], i2 = tidx[3*f+2];
                float hx = u*verts[3*i0  ] + v*verts[3*i1  ] + w*verts[3*i2  ];
                float hy = u*verts[3*i0+1] + v*verts[3*i1+1] + w*verts[3*i2+1];
                float hz = u*verts[3*i0+2] + v*verts[3*i1+2] + w*verts[3*i2+2];
                float dx = px[s]-hx, dy = py[s]-hy, dz = pz[s]-hz;
                float dist = sqrtf(dx*dx + dy*dy + dz*dz);
                float winding = om * 0.07957747154594766788f; // 1/(4*pi)
                float sgn = (winding > 0.5f) ? -1.f : 1.f;
                out[gp] = sgn * dist;
                float* hitp = out + nPts;
                hitp[3*gp  ] = hx;
                hitp[3*gp+1] = hy;
                hitp[3*gp+2] = hz;
                ((int*)(out + (size_t)nPts * 4))[gp] = f;
            }
        }
    }
}

extern "C" void kernel_launch(void* const* d_in, const int* in_sizes, int n_in,
                              void* d_out, int out_size, void* d_ws, size_t ws_size,
                              hipStream_t stream) {
    const float* points = (const float*)d_in[0];
    const float* verts  = (const float*)d_in[1];
    const int*   tidx   = (const int*)d_in[2];
    float* out = (float*)d_out;
    const int nPts = in_sizes[0] / 3;   // 8192
    const int nTri = in_sizes[2] / 3;   // 2048
    const int blocks = (nPts + 127) / 128;  // 128 points per 256-thread block
    sdf_wmma_kernel<<<blocks, BLOCK, 0, stream>>>(points, verts, tidx, out, nPts, nTri);
}